// NonMaxSuppression_3470333575867
// MI455X (gfx1250) — compile-verified
//
#include <hip/hip_runtime.h>
#include <hip/hip_bf16.h>

typedef __attribute__((ext_vector_type(2))) float v2f;
typedef __attribute__((ext_vector_type(8))) float v8f;

#define NBOX   1024
#define NIMG   4
#define NCLS   80
#define KPC    100
#define MAXT   300
#define CAND   (NCLS * KPC)      /* 8000 */
#define MWORDS (NBOX / 32)       /* 32 */
#define NEGV   (-1e30f)
#define IOU_T  0.5f
#define SCORE_T 0.001f
#define EPSV   1e-12f

// ---------------------------------------------------------------------------
// Kernel 1: per-image pairwise IoU suppression bitmask (class independent!)
// One wave (32 threads) per (image, 16-row tile). The union pre-term
// (area_i + area_j) for a 16x16 tile is a rank-2 outer product computed on the
// matrix pipe via V_WMMA_F32_16X16X4_F32; intersections on the VALU; bits
// packed with wave32 ballots.
// ---------------------------------------------------------------------------
__global__ __launch_bounds__(32)
void iou_mask_kernel(const float* __restrict__ boxes, unsigned* __restrict__ mask) {
    const int blk  = blockIdx.x;        // NIMG * 64 blocks
    const int img  = blk >> 6;
    const int m0   = (blk & 63) * 16;   // first row of this tile
    const int lane = threadIdx.x;

    const float* bimg = boxes + (size_t)img * NBOX * 4;

    // A-matrix (16x4 f32, 2 VGPRs): K=0 -> row areas, K=1 -> ones, K=2/3 -> 0
    const int arow = m0 + (lane & 15);
    float r0 = bimg[arow * 4 + 0], r1 = bimg[arow * 4 + 1];
    float r2 = bimg[arow * 4 + 2], r3 = bimg[arow * 4 + 3];
    float ay1 = fminf(r0, r2), ay2 = fmaxf(r0, r2);
    float ax1 = fminf(r1, r3), ax2 = fmaxf(r1, r3);
    const float areaR = (ay2 - ay1) * (ax2 - ax1);

    v2f A;
    A.x = (lane < 16) ? areaR : 0.0f;   // lanes 0-15: K=0 ; lanes 16-31: K=2 (pad)
    A.y = (lane < 16) ? 1.0f  : 0.0f;   // lanes 0-15: K=1 ; lanes 16-31: K=3 (pad)

    // Canonicalized row boxes this lane needs for intersections:
    // lanes 0-15 handle rows m0..m0+7 (C VGPR r), lanes 16-31 rows m0+8..m0+15.
    float by1[8], bx1[8], by2[8], bx2[8];
    const int rbase = m0 + ((lane >= 16) ? 8 : 0);
#pragma unroll
    for (int r = 0; r < 8; ++r) {
        float q0 = bimg[(rbase + r) * 4 + 0], q1 = bimg[(rbase + r) * 4 + 1];
        float q2 = bimg[(rbase + r) * 4 + 2], q3 = bimg[(rbase + r) * 4 + 3];
        by1[r] = fminf(q0, q2); by2[r] = fmaxf(q0, q2);
        bx1[r] = fminf(q1, q3); bx2[r] = fmaxf(q1, q3);
    }

    // lane < 16 owns output row (m0 + lane); accumulates 16-bit tile halves.
    const unsigned rowbase = ((unsigned)img * NBOX + (unsigned)(m0 + lane)) * MWORDS;
    unsigned pend = 0;

    for (int c = 0; c < 64; ++c) {
        const int col = c * 16 + (lane & 15);
        float c0 = bimg[col * 4 + 0], c1 = bimg[col * 4 + 1];
        float c2 = bimg[col * 4 + 2], c3 = bimg[col * 4 + 3];
        float cy1 = fminf(c0, c2), cy2 = fmaxf(c0, c2);
        float cx1 = fminf(c1, c3), cx2 = fmaxf(c1, c3);
        const float areaC = (cy2 - cy1) * (cx2 - cx1);

        // B-matrix (4x16 f32): K=0 row of ones, K=1 row of col areas, K=2/3 pad
        v2f Bv;
        Bv.x = (lane < 16) ? 1.0f  : 0.0f;
        Bv.y = (lane < 16) ? areaC : 0.0f;

        v8f Dz = {};
        // D[m][n] = area_row[m] + area_col[n]  (rank-2 outer product on WMMA)
        v8f D = __builtin_amdgcn_wmma_f32_16x16x4_f32(
            false, A, false, Bv, (short)0, Dz, false, false);

        unsigned bits = 0;
#pragma unroll
        for (int r = 0; r < 8; ++r) {
            float ih = fmaxf(fminf(by2[r], cy2) - fmaxf(by1[r], cy1), 0.0f);
            float iw = fmaxf(fminf(bx2[r], cx2) - fmaxf(bx1[r], cx1), 0.0f);
            float inter = ih * iw;
            float uni = fmaxf(D[r] - inter, EPSV);
            bool hit = inter > IOU_T * uni;               // iou > 0.5
            unsigned bal = __builtin_amdgcn_ballot_w32(hit);
            // ballot: lanes 0-15 -> row r (cols 0-15), lanes 16-31 -> row r+8
            if (lane == r)     bits = bal & 0xFFFFu;
            if (lane == r + 8) bits = bal >> 16;
        }
        if ((c & 1) == 0) {
            pend = bits;
        } else if (lane < 16) {
            mask[rowbase + (c >> 1)] = pend | (bits << 16);
        }
    }
}

// ---------------------------------------------------------------------------
// Kernel 2: per (image,class) NMS. 256 threads: bitonic sort 1024 candidates
// in LDS (key desc, index asc tiebreak == stable argsort(-s)), then wave 0
// runs the greedy scan: lane k owns kept-bitmask word k, wave ballot gives
// "any previously-kept overlap" from the shared suppression matrix.
// ---------------------------------------------------------------------------
__global__ __launch_bounds__(256)
void per_class_nms(const float* __restrict__ scores,
                   const float* __restrict__ boxes,
                   const unsigned* __restrict__ mask,
                   float* __restrict__ cand_score,
                   float* __restrict__ cand_box) {
    __shared__ float          skey[NBOX];
    __shared__ unsigned short sidx[NBOX];

    const int blk = blockIdx.x;          // NIMG * NCLS
    const int img = blk / NCLS;
    const int cls = blk % NCLS;
    const int tid = threadIdx.x;

    const float* simg = scores + (size_t)img * NBOX * NCLS + cls;
    for (int i = tid; i < NBOX; i += 256) {
        float s = simg[(size_t)i * NCLS];
        skey[i] = (s > SCORE_T) ? s : NEGV;
        sidx[i] = (unsigned short)i;
    }
    __syncthreads();

    // bitonic sort, descending by key, ascending index tiebreak
    for (int k = 2; k <= NBOX; k <<= 1) {
        for (int j = k >> 1; j > 0; j >>= 1) {
            for (int i = tid; i < NBOX; i += 256) {
                int l = i ^ j;
                if (l > i) {
                    float ki = skey[i], kl = skey[l];
                    unsigned short ii = sidx[i], il = sidx[l];
                    bool i_first = (ki > kl) || (ki == kl && ii < il);
                    bool desc = ((i & k) == 0);
                    if (desc ? !i_first : i_first) {
                        skey[i] = kl; skey[l] = ki;
                        sidx[i] = il; sidx[l] = ii;
                    }
                }
            }
            __syncthreads();
        }
    }

    if (tid < 32) {
        const int lane = tid;
        unsigned kept = 0;               // lane owns 32 of 1024 kept bits
        int cnt = 0;
        const int base = (img * NCLS + cls) * KPC;
        const float* bimg = boxes + (size_t)img * NBOX * 4;
        const unsigned* mimg = mask + (size_t)img * NBOX * MWORDS;

        for (int i = 0; i < NBOX; ++i) {
            float s = skey[i];
            if (!(s > -1e29f)) break;    // sorted: rest invalid
            int idx = (int)sidx[i];
            if (i + 1 < NBOX)
                __builtin_prefetch(mimg + (size_t)sidx[i + 1] * MWORDS + lane, 0, 0);
            unsigned row = mimg[(size_t)idx * MWORDS + lane];
            bool sup = __builtin_amdgcn_ballot_w32((kept & row) != 0u) != 0u;
            if (!sup) {
                if (lane == (idx >> 5)) kept |= 1u << (idx & 31);
                int slot = base + cnt;
                if (lane == 0) cand_score[slot] = s;
                if (lane < 4)  cand_box[(size_t)slot * 4 + lane] = bimg[(size_t)idx * 4 + lane];
                ++cnt;
                if (cnt == KPC) break;
            }
        }
        for (int j = cnt + lane; j < KPC; j += 32) cand_score[base + j] = NEGV;
    }
}

// ---------------------------------------------------------------------------
// Kernel 3: per image top-300 of 8000 candidates (bitonic sort of 8192 in
// LDS), clip boxes, emit (scores, boxes, classes, valid_count).
// ---------------------------------------------------------------------------
#define NS 8192
__global__ __launch_bounds__(512)
void topk_final(const float* __restrict__ cand_score,
                const float* __restrict__ cand_box,
                float* __restrict__ out) {
    __shared__ float          skey[NS];
    __shared__ unsigned short sidx[NS];
    __shared__ int cntS;

    const int img = blockIdx.x;
    const int tid = threadIdx.x;
    const float* cs = cand_score + (size_t)img * CAND;

    if (tid == 0) cntS = 0;
    for (int i = tid; i < NS; i += 512) {
        skey[i] = (i < CAND) ? cs[i] : NEGV;   // pads lose ties via larger index
        sidx[i] = (unsigned short)i;
    }
    __syncthreads();

    for (int k = 2; k <= NS; k <<= 1) {
        for (int j = k >> 1; j > 0; j >>= 1) {
            for (int i = tid; i < NS; i += 512) {
                int l = i ^ j;
                if (l > i) {
                    float ki = skey[i], kl = skey[l];
                    unsigned short ii = sidx[i], il = sidx[l];
                    bool i_first = (ki > kl) || (ki == kl && ii < il);
                    bool desc = ((i & k) == 0);
                    if (desc ? !i_first : i_first) {
                        skey[i] = kl; skey[l] = ki;
                        sidx[i] = il; sidx[l] = ii;
                    }
                }
            }
            __syncthreads();
        }
    }

    if (tid < MAXT) {
        float kv = skey[tid];
        int ci = (int)sidx[tid];
        bool v = kv > -0.5e30f;                 // top_s > NEG/2
        int cis = ci < (CAND - 1) ? ci : (CAND - 1);  // clamp (pads) for safe load

        out[(size_t)img * MAXT + tid] = v ? kv : 0.0f;

        const float* bp = cand_box + ((size_t)img * CAND + cis) * 4;
#pragma unroll
        for (int jdim = 0; jdim < 4; ++jdim) {
            float bv = 0.0f;
            if (v) bv = fminf(fmaxf(bp[jdim], 0.0f), 1.0f);   // clip_boxes
            out[(size_t)NIMG * MAXT + ((size_t)img * MAXT + tid) * 4 + jdim] = bv;
        }
        out[(size_t)NIMG * MAXT * 5 + (size_t)img * MAXT + tid] =
            v ? (float)(ci / KPC) : 0.0f;
        if (v) atomicAdd(&cntS, 1);
    }
    __syncthreads();
    if (tid == 0) out[(size_t)NIMG * MAXT * 6 + img] = (float)cntS;
}

// ---------------------------------------------------------------------------
extern "C" void kernel_launch(void* const* d_in, const int* in_sizes, int n_in,
                              void* d_out, int out_size, void* d_ws, size_t ws_size,
                              hipStream_t stream) {
    const float* scores = (const float*)d_in[0];   // [4,1024,80]
    const float* boxes  = (const float*)d_in[1];   // [4,1024,1,4]
    float* out = (float*)d_out;                    // 1200 + 4800 + 1200 + 4

    char* ws = (char*)d_ws;
    unsigned* mask    = (unsigned*)ws;                                   // 512 KB
    float* cand_score = (float*)(ws + (size_t)NIMG * NBOX * MWORDS * 4); // 128 KB
    float* cand_box   = cand_score + (size_t)NIMG * CAND;                // 512 KB

    iou_mask_kernel<<<NIMG * 64, 32, 0, stream>>>(boxes, mask);
    per_class_nms<<<NIMG * NCLS, 256, 0, stream>>>(scores, boxes, mask,
                                                   cand_score, cand_box);
    topk_final<<<NIMG, 512, 0, stream>>>(cand_score, cand_box, out);
}